// LeNet_vis_32787780338220
// MI455X (gfx1250) — compile-verified
//
#include <hip/hip_runtime.h>
#include <hip/hip_bf16.h>

// ---------- CDNA5 helpers: async global->LDS copy (ASYNCcnt path) ----------
typedef int v4i_vs __attribute__((vector_size(16)));
typedef __attribute__((address_space(1))) v4i_vs* gv4i_p;  // global (AS1)
typedef __attribute__((address_space(3))) v4i_vs* lv4i_p;  // LDS (AS3)

__device__ __forceinline__ void async_copy16(const float* g, float* l) {
#if __has_builtin(__builtin_amdgcn_global_load_async_to_lds_b128)
  __builtin_amdgcn_global_load_async_to_lds_b128((gv4i_p)g, (lv4i_p)l, 0, 0);
#else
  *(float4*)l = *(const float4*)g;
#endif
}

__device__ __forceinline__ void wait_async0() {
#if __has_builtin(__builtin_amdgcn_s_wait_asynccnt)
  __builtin_amdgcn_s_wait_asynccnt(0);
#elif __has_builtin(__builtin_amdgcn_global_load_async_to_lds_b128)
  asm volatile("s_wait_asynccnt 0" ::: "memory");
#endif
}

// =====================================================================
// Kernel 1: adder conv1 (C=1 -> O=32, 5x5, pad 2) + PReLU(a1)
// grid = 256 (one block per batch image), block = 256
// =====================================================================
__global__ void adder1_kernel(const float* __restrict__ x,
                              const float* __restrict__ w1,
                              const float* __restrict__ a1,
                              float* __restrict__ h1) {
  __shared__ float xb[784];    // 28*28 input image
  __shared__ float wsh[800];   // 32*25 weights
  const int tid = threadIdx.x;
  const int b = blockIdx.x;

  if (tid < 196) async_copy16(x + (size_t)b * 784 + tid * 4, &xb[tid * 4]);
  if (tid < 200) async_copy16(w1 + tid * 4, &wsh[tid * 4]);
  wait_async0();
  __syncthreads();

  const float a1v = a1[0];
  for (int o = 0; o < 32; ++o) {
    const float* wo = &wsh[o * 25];
    for (int p = tid; p < 784; p += 256) {
      const int i = p / 28, j = p % 28;
      float s = 0.f;
#pragma unroll
      for (int kh = 0; kh < 5; ++kh) {
        const int ii = i + kh - 2;
#pragma unroll
        for (int kw = 0; kw < 5; ++kw) {
          const int jj = j + kw - 2;
          const bool ok = ((unsigned)ii < 28u) && ((unsigned)jj < 28u);
          float xv = xb[ok ? ii * 28 + jj : 0];
          xv = ok ? xv : 0.f;
          s += __builtin_fabsf(xv - wo[kh * 5 + kw]);
        }
      }
      const float v = -s;
      h1[(size_t)b * 25088 + o * 784 + p] = (v >= 0.f) ? v : a1v * v;
    }
  }
}

// =====================================================================
// Kernel 2: adder conv2 (32->32, 5x5, pad 2) + PReLU(a2) + fused 3x maxpool
// Only the 24x24 region feeding the pool is computed; h2 never hits memory.
// grid = 256*4 (block = (b, og) with 8 output channels per og), block = 288
// =====================================================================
__global__ void adder2_pool_kernel(const float* __restrict__ h1,
                                   const float* __restrict__ w2,
                                   const float* __restrict__ a2,
                                   float* __restrict__ pooled) {
  __shared__ float xs[8 * 784];    // 8 input channels of 28x28   (25.0 KB)
  __shared__ float wsh[8 * 800];   // 8 filters x 800 weights     (25.6 KB)
  __shared__ float part[8 * 144];  // 2x2-max grid (12x12) per o  ( 4.6 KB)

  const int tid = threadIdx.x;          // 0..287
  const int b = blockIdx.x >> 2;
  const int og = blockIdx.x & 3;

  // Filters for this output-channel group (contiguous 6400 floats).
  for (int k = tid; k < 1600; k += 288)
    async_copy16(w2 + (size_t)og * 6400 + k * 4, &wsh[k * 4]);

  float acc[4][4];
#pragma unroll
  for (int it = 0; it < 4; ++it)
#pragma unroll
    for (int p = 0; p < 4; ++p) acc[it][p] = 0.f;

  // Stream input channels in chunks of 8 through LDS.
  for (int cc = 0; cc < 4; ++cc) {
    if (cc) __syncthreads();  // previous chunk fully consumed
    for (int k = tid; k < 1568; k += 288)
      async_copy16(h1 + (size_t)b * 25088 + cc * 6272 + k * 4, &xs[k * 4]);
    wait_async0();
    __syncthreads();

#pragma unroll
    for (int it = 0; it < 4; ++it) {
      const int t = tid + it * 288;            // 0..1151
      const int o = t / 144, q = t % 144;      // o: local out channel
      const int i0 = (q / 12) * 2, j0 = (q % 12) * 2;
      const float* wb = &wsh[o * 800 + cc * 200];
      for (int c = 0; c < 8; ++c) {
        const float* xc = &xs[c * 784];
        const float* wc = wb + c * 25;
#pragma unroll
        for (int kh = 0; kh < 5; ++kh) {
#pragma unroll
          for (int kw = 0; kw < 5; ++kw) {
            const float w = wc[kh * 5 + kw];
#pragma unroll
            for (int p = 0; p < 4; ++p) {
              const int ii = i0 + (p >> 1) + kh - 2;  // <= 25, only low check
              const int jj = j0 + (p & 1) + kw - 2;
              const bool ok = (ii >= 0) && (jj >= 0);
              float xv = xc[ok ? ii * 28 + jj : 0];
              xv = ok ? xv : 0.f;
              acc[it][p] += __builtin_fabsf(xv - w);
            }
          }
        }
      }
    }
  }

  // PReLU + 2x2 max -> part
  const float a2v = a2[0];
#pragma unroll
  for (int it = 0; it < 4; ++it) {
    const int t = tid + it * 288;
    const int o = t / 144, q = t % 144;
    float m = -__builtin_inff();
#pragma unroll
    for (int p = 0; p < 4; ++p) {
      float v = -acc[it][p];
      v = (v >= 0.f) ? v : a2v * v;
      m = fmaxf(m, v);
    }
    part[o * 144 + q] = m;
  }
  __syncthreads();

  // 4x4 reduction of part -> one pooled value (8x8 window of pixels)
  if (tid < 72) {
    const int o = tid / 9, r = tid % 9, pi = r / 3, pj = r % 3;
    float m = -__builtin_inff();
#pragma unroll
    for (int a = 0; a < 4; ++a)
#pragma unroll
      for (int c = 0; c < 4; ++c)
        m = fmaxf(m, part[o * 144 + (pi * 4 + a) * 12 + pj * 4 + c]);
    pooled[(size_t)b * 288 + (og * 8 + o) * 9 + pi * 3 + pj] = m;
  }
}

// =====================================================================
// Kernel 3: FC1 via V_WMMA_F32_16X16X4_F32:  [256,288] @ [288,2]^T + b, PReLU
// grid = 16 (one wave per 16-row tile), block = 32 (one wave)
// B lanes with N >= 2 read a safe (row 0) address and mask the value to 0,
// keeping the K-loop branch-free with EXEC all-ones end to end.
// =====================================================================
typedef __attribute__((ext_vector_type(2))) float v2f;
typedef __attribute__((ext_vector_type(8))) float v8f;

__global__ void fc1_wmma_kernel(const float* __restrict__ flat,
                                const float* __restrict__ w,
                                const float* __restrict__ bias,
                                const float* __restrict__ a3,
                                float* __restrict__ ip1) {
  const int lane = threadIdx.x;          // 0..31 (wave32)
  const int row0 = blockIdx.x * 16;
  const int m = lane & 15;               // A row / B col (N)
  const int g = lane >> 4;               // K-half selector
  const float* arow = flat + (size_t)(row0 + m) * 288 + 2 * g;
  const int bn = (m < 2) ? m : 0;                    // clamp to a valid row
  const float* brow = w + (size_t)bn * 288 + 2 * g;  // always-safe address
  const float bmask = (m < 2) ? 1.f : 0.f;           // zero pad columns 2..15

  v8f c = {0.f, 0.f, 0.f, 0.f, 0.f, 0.f, 0.f, 0.f};
  for (int k0 = 0; k0 < 288; k0 += 4) {
    v2f a, bb;
    a.x = arow[k0];
    a.y = arow[k0 + 1];
    bb.x = brow[k0] * bmask;
    bb.y = brow[k0 + 1] * bmask;
    // D = A(16x4) x B(4x16) + C  — fp32 WMMA
    c = __builtin_amdgcn_wmma_f32_16x16x4_f32(false, a, false, bb, (short)0, c,
                                              false, false);
  }

  const float a3v = a3[0];
  if (m < 2) {  // N = m; only 2 valid output columns
    const float bv = bias[m];
#pragma unroll
    for (int r = 0; r < 8; ++r) {
      const int M = r + 8 * g;
      float v = c[r] + bv;
      v = (v >= 0.f) ? v : a3v * v;
      ip1[(size_t)(row0 + M) * 2 + m] = v;
    }
  }
}

// =====================================================================
// Kernel 4: FC2  [256,2] @ [2,10]^T  (trivial)
// =====================================================================
__global__ void fc2_kernel(const float* __restrict__ ip1,
                           const float* __restrict__ w,
                           float* __restrict__ ip2) {
  const int idx = blockIdx.x * blockDim.x + threadIdx.x;
  if (idx < 2560) {
    const int b = idx / 10, n = idx % 10;
    ip2[idx] = ip1[b * 2 + 0] * w[n * 2 + 0] + ip1[b * 2 + 1] * w[n * 2 + 1];
  }
}

// =====================================================================
extern "C" void kernel_launch(void* const* d_in, const int* in_sizes, int n_in,
                              void* d_out, int out_size, void* d_ws,
                              size_t ws_size, hipStream_t stream) {
  (void)in_sizes; (void)n_in; (void)out_size; (void)ws_size;
  // setup_inputs order: x, w1, a1, w2, a2, ip1_w, ip1_b, a3, ip2_w
  const float* x    = (const float*)d_in[0];
  const float* w1   = (const float*)d_in[1];
  const float* a1   = (const float*)d_in[2];
  const float* w2   = (const float*)d_in[3];
  const float* a2   = (const float*)d_in[4];
  const float* ip1w = (const float*)d_in[5];
  const float* ip1b = (const float*)d_in[6];
  const float* a3   = (const float*)d_in[7];
  const float* ip2w = (const float*)d_in[8];

  float* out = (float*)d_out;            // [0,512) = ip1, [512,3072) = ip2
  float* h1 = (float*)d_ws;              // 256*32*28*28 floats (25.7 MB)
  float* pooled = h1 + (size_t)25088 * 256;  // 256*288 floats

  adder1_kernel<<<256, 256, 0, stream>>>(x, w1, a1, h1);
  adder2_pool_kernel<<<1024, 288, 0, stream>>>(h1, w2, a2, pooled);
  fc1_wmma_kernel<<<16, 32, 0, stream>>>(pooled, ip1w, ip1b, a3, out);
  fc2_kernel<<<10, 256, 0, stream>>>(out, ip2w, out + 512);
}